// FCResNetBlock_6390911336492
// MI455X (gfx1250) — compile-verified
//
#include <hip/hip_runtime.h>
#include <math.h>

// FieldConv ResNet block for MI455X (gfx1250, wave32).
//
//   g[n,rm,d] = sum_c x[n,c] * Wc[rm,c,d]          (WMMA f32 16x16x4, weights in LDS)
//   t_e[d]    = sum_rm stenc[e,rm] * g[src_e,rm,d] (per-edge wave; scalar stencil loads)
//   h[n,d]    = atomic f32 segment-sum over tgt    (256B atomics/edge)
//   nonlin, conv2, residual WMMA GEMM + nonlin -> out.
//
// g (46MB) is L2-resident (192MB L2); HBM traffic ~100MB -> microseconds at 23.3TB/s.
// fp32 WMMA on purpose: the op is L2-gather/atomic bound, lower precision buys nothing.

#define CINC 32
#define COUTC 32
#define RMT 18          // R*M = 6*3
#define EPSV 1e-8f

typedef __attribute__((ext_vector_type(2))) float v2f;
typedef __attribute__((ext_vector_type(8))) float v8f;

__device__ __forceinline__ v8f wmma_f32(v2f a, v2f b, v8f c) {
  // V_WMMA_F32_16X16X4_F32: D = A(16x4) * B(4x16) + C(16x16), fp32.
  return __builtin_amdgcn_wmma_f32_16x16x4_f32(
      /*neg_a=*/false, a, /*neg_b=*/false, b,
      /*c_mod=*/(short)0, c, /*reuse_a=*/false, /*reuse_b=*/false);
}

// A fragments for one K-step: one 16B load gives re+im for K=kk,kk+1.
// Layout: lane<16 -> A[row][k0..k0+1], lane>=16 -> A[row][k0+2..k0+3].
__device__ __forceinline__ void load_a_pair(const float2* __restrict__ x, int n0,
                                            int k0, int lane, int N,
                                            v2f& are, v2f& aim) {
  int row = n0 + (lane & 15);
  if (row >= N) row = N - 1;                   // tail clamp; masked at store
  int kk = k0 + ((lane >> 4) << 1);            // kk even -> 16B aligned
  float4 e = *(const float4*)&x[row * CINC + kk];
  are.x = e.x; are.y = e.z;                    // re of K=kk, kk+1
  aim.x = e.y; aim.y = e.w;                    // im of K=kk, kk+1
}

// B fragment (4x16 f32) from a pair-swizzled LDS plane:
//   sW[(k>>1)*64 + col*2 + (k&1)] = W[k][col]  -> one ds_load_b64 per fragment.
__device__ __forceinline__ v2f load_b_lds(const float* __restrict__ sW, int k0,
                                          int col0, int lane) {
  int col = col0 + (lane & 15);
  int kb = k0 + ((lane >> 4) << 1);            // kb even
  return *(const v2f*)&sW[(kb >> 1) * 64 + col * 2];
}

// ---------------- kernels ----------------

__global__ void zero_f2_kernel(float2* __restrict__ p, int n) {
  int i = blockIdx.x * blockDim.x + threadIdx.x;
  if (i < n) p[i] = make_float2(0.f, 0.f);
}

// Wc[rm,c,d] = w[rm,c,d] * exp(i*off[c,d]) split into planar re/im.
__global__ void wc_kernel(const float* __restrict__ w, const float* __restrict__ off,
                          float* __restrict__ Wre, float* __restrict__ Wim) {
  int i = blockIdx.x * blockDim.x + threadIdx.x;
  if (i >= RMT * CINC * COUTC) return;
  int cd = i & (CINC * COUTC - 1);
  float s, c;
  sincosf(off[cd], &s, &c);
  float wv = w[i];
  Wre[i] = wv * c;
  Wim[i] = wv * s;
}

// g[n, j, d] (complex) = sum_c xin[n,c] * Wc[j][c,d].
// One wave per 16-node tile; all 4 waves of a block share weight plane j via LDS.
__global__ void __launch_bounds__(128)
g_kernel(const float2* __restrict__ xin,
         const float* __restrict__ Wre, const float* __restrict__ Wim,
         float2* __restrict__ g, int cw, int numTiles, int N) {
  __shared__ float sWre[CINC * COUTC];
  __shared__ float sWim[CINC * COUTC];
  int j = blockIdx.y;
  {
    const float* wre = Wre + j * (CINC * COUTC);
    const float* wim = Wim + j * (CINC * COUTC);
    for (int i = threadIdx.x; i < CINC * COUTC; i += 128) {
      int k = i >> 5, col = i & 31;
      int d = (k >> 1) * 64 + col * 2 + (k & 1);   // pair-swizzle for b64 fragments
      sWre[d] = wre[i];
      sWim[d] = wim[i];
    }
  }
  __syncthreads();                               // before any early-return!

  int lane = threadIdx.x & 31;
  int tile = blockIdx.x * (blockDim.x >> 5) + (threadIdx.x >> 5);
  if (tile >= numTiles) return;                  // wave-uniform; EXEC stays full
  int n0 = tile * 16;

  v8f rr0 = {}, rr1 = {}, ii0 = {}, ii1 = {}, mi0 = {}, mi1 = {};
  for (int k0 = 0; k0 < CINC; k0 += 4) {
    v2f are, aim;
    load_a_pair(xin, n0, k0, lane, N, are, aim);
    v2f br0 = load_b_lds(sWre, k0, 0, lane);
    v2f br1 = load_b_lds(sWre, k0, 16, lane);
    v2f bi0 = load_b_lds(sWim, k0, 0, lane);
    v2f bi1 = load_b_lds(sWim, k0, 16, lane);
    rr0 = wmma_f32(are, br0, rr0);               // re*We
    rr1 = wmma_f32(are, br1, rr1);
    ii0 = wmma_f32(aim, bi0, ii0);               // im*Wi
    ii1 = wmma_f32(aim, bi1, ii1);
    mi0 = wmma_f32(are, bi0, mi0);               // re*Wi + im*We
    mi0 = wmma_f32(aim, br0, mi0);
    mi1 = wmma_f32(are, bi1, mi1);
    mi1 = wmma_f32(aim, br1, mi1);
  }
  int d0 = lane & 15;
  for (int i = 0; i < 8; ++i) {
    int n = n0 + i + ((lane >> 4) << 3);
    if (n >= N) continue;
    size_t base = ((size_t)n * cw + j) * COUTC;
    g[base + d0]      = make_float2(rr0[i] - ii0[i], mi0[i]);
    g[base + d0 + 16] = make_float2(rr1[i] - ii1[i], mi1[i]);
  }
}

// One wave per edge; lane owns output channel d. Edge indices and stencil taps are
// wave-uniform -> force scalar (SMEM/KMcnt) loads via readfirstlane so the vector
// pipe only carries the coalesced 256B g[src] gather and the output atomics.
__global__ void edge_kernel(const int* __restrict__ edges, const float2* __restrict__ sten,
                            const float2* __restrict__ g, float* __restrict__ acc,
                            int E, int cw, int rmBase) {
  int wid = __builtin_amdgcn_readfirstlane((blockIdx.x * blockDim.x + threadIdx.x) >> 5);
  int lane = threadIdx.x & 31;
  if (wid >= E) return;                          // wave-uniform
  int src = edges[2 * wid];                      // scalar load
  int tgt = edges[2 * wid + 1];
  const float2* se = sten + (size_t)wid * RMT + rmBase;
  const float2* gs = g + (size_t)src * cw * COUTC + lane;
  float tre = 0.f, tim = 0.f;
  for (int j = 0; j < cw; ++j) {
    float2 s = se[j];                            // scalar (uniform) load
    float2 gv = gs[(size_t)j * COUTC];           // 256B coalesced per wave
    tre = fmaf(s.x, gv.x, fmaf(-s.y, gv.y, tre));
    tim = fmaf(s.x, gv.y, fmaf(s.y, gv.x, tim));
  }
  float* a = acc + ((size_t)tgt * COUTC + lane) * 2;
  atomicAdd(a, tre);
  atomicAdd(a + 1, tim);
}

// h = relu(|z|+b)/max(|z|,eps) * z
__global__ void nonlin_kernel(const float2* __restrict__ acc, const float* __restrict__ b,
                              float2* __restrict__ out, int total) {
  int i = blockIdx.x * blockDim.x + threadIdx.x;
  if (i >= total) return;
  float2 z = acc[i];
  float mag = sqrtf(z.x * z.x + z.y * z.y);
  float sc = fmaxf(mag + b[i & (COUTC - 1)], 0.f) / fmaxf(mag, EPSV);
  out[i] = make_float2(z.x * sc, z.y * sc);
}

// y = nonlin(x @ res_w + h2, b2); residual complex GEMM via WMMA, res_w staged in LDS.
__global__ void __launch_bounds__(128)
final_kernel(const float2* __restrict__ x, const float2* __restrict__ resw,
             const float2* __restrict__ h2, const float* __restrict__ b2,
             float2* __restrict__ out, int numTiles, int N) {
  __shared__ float sWre[CINC * COUTC];
  __shared__ float sWim[CINC * COUTC];
  for (int i = threadIdx.x; i < CINC * COUTC; i += 128) {
    int k = i >> 5, col = i & 31;
    int d = (k >> 1) * 64 + col * 2 + (k & 1);
    float2 e = resw[i];                          // interleaved complex (Cin,Cout)
    sWre[d] = e.x;
    sWim[d] = e.y;
  }
  __syncthreads();

  int lane = threadIdx.x & 31;
  int tile = blockIdx.x * (blockDim.x >> 5) + (threadIdx.x >> 5);
  if (tile >= numTiles) return;
  int n0 = tile * 16;

  v8f rr0 = {}, rr1 = {}, ii0 = {}, ii1 = {}, mi0 = {}, mi1 = {};
  for (int k0 = 0; k0 < CINC; k0 += 4) {
    v2f are, aim;
    load_a_pair(x, n0, k0, lane, N, are, aim);
    v2f br0 = load_b_lds(sWre, k0, 0, lane);
    v2f br1 = load_b_lds(sWre, k0, 16, lane);
    v2f bi0 = load_b_lds(sWim, k0, 0, lane);
    v2f bi1 = load_b_lds(sWim, k0, 16, lane);
    rr0 = wmma_f32(are, br0, rr0);
    rr1 = wmma_f32(are, br1, rr1);
    ii0 = wmma_f32(aim, bi0, ii0);
    ii1 = wmma_f32(aim, bi1, ii1);
    mi0 = wmma_f32(are, bi0, mi0);
    mi0 = wmma_f32(aim, br0, mi0);
    mi1 = wmma_f32(are, bi1, mi1);
    mi1 = wmma_f32(aim, br1, mi1);
  }
  int d0 = lane & 15;
  for (int i = 0; i < 8; ++i) {
    int n = n0 + i + ((lane >> 4) << 3);
    if (n >= N) continue;
    for (int nt = 0; nt < 2; ++nt) {
      int d = d0 + nt * 16;
      float re = (nt ? rr1[i] - ii1[i] : rr0[i] - ii0[i]);
      float im = (nt ? mi1[i] : mi0[i]);
      float2 h = h2[(size_t)n * COUTC + d];
      re += h.x;
      im += h.y;
      float mag = sqrtf(re * re + im * im);
      float sc = fmaxf(mag + b2[d], 0.f) / fmaxf(mag, EPSV);
      out[(size_t)n * COUTC + d] = make_float2(re * sc, im * sc);
    }
  }
}

// ---------------- launch ----------------

extern "C" void kernel_launch(void* const* d_in, const int* in_sizes, int n_in,
                              void* d_out, int out_size, void* d_ws, size_t ws_size,
                              hipStream_t stream) {
  const float2* x2    = (const float2*)d_in[0];
  const int*    edges = (const int*)d_in[1];
  const float2* sten2 = (const float2*)d_in[2];
  const float*  w1    = (const float*)d_in[3];
  const float*  off1  = (const float*)d_in[4];
  const float*  b1    = (const float*)d_in[5];
  const float*  w2    = (const float*)d_in[6];
  const float*  off2  = (const float*)d_in[7];
  const float*  b2    = (const float*)d_in[8];
  const float2* resw2 = (const float2*)d_in[9];

  const int N = in_sizes[0] / (CINC * 2);
  const int E = in_sizes[1] / 2;
  const int numTiles = (N + 15) / 16;

  // Workspace: [Wre | Wim | acc (N,32,cplx) | hbuf (N,32,cplx) | g (N,CH,32,cplx)]
  char* wsb = (char*)d_ws;
  float* Wre  = (float*)wsb;
  float* Wim  = Wre + RMT * CINC * COUTC;
  float* accF = Wim + RMT * CINC * COUTC;
  float2* hbuf = (float2*)(accF + (size_t)N * COUTC * 2);
  float2* g    = hbuf + (size_t)N * COUTC;

  size_t fixedB = (size_t)2 * RMT * CINC * COUTC * sizeof(float)
                + (size_t)2 * N * COUTC * sizeof(float2);
  size_t perRm  = (size_t)N * COUTC * sizeof(float2);
  int CH = RMT;
  if (ws_size > fixedB) {
    size_t fit = (ws_size - fixedB) / perRm;
    if (fit < (size_t)CH) CH = (int)fit;
  } else {
    CH = 1;
  }
  if (CH < 1) CH = 1;

  dim3 blk256(256), blk128(128);

  for (int conv = 0; conv < 2; ++conv) {
    const float*  w   = conv == 0 ? w1 : w2;
    const float*  off = conv == 0 ? off1 : off2;
    const float2* xin = conv == 0 ? x2 : (const float2*)hbuf;

    wc_kernel<<<(RMT * CINC * COUTC + 255) / 256, blk256, 0, stream>>>(w, off, Wre, Wim);
    zero_f2_kernel<<<((size_t)N * COUTC + 255) / 256, blk256, 0, stream>>>((float2*)accF, N * COUTC);

    for (int cs = 0; cs < RMT; cs += CH) {
      int cw = (RMT - cs) < CH ? (RMT - cs) : CH;
      dim3 ggrid((numTiles + 3) / 4, cw);
      g_kernel<<<ggrid, blk128, 0, stream>>>(xin, Wre + cs * CINC * COUTC,
                                             Wim + cs * CINC * COUTC, g, cw, numTiles, N);
      edge_kernel<<<((size_t)E * 32 + 255) / 256, blk256, 0, stream>>>(
          edges, sten2, g, accF, E, cw, cs);
    }

    if (conv == 0) {
      nonlin_kernel<<<((size_t)N * COUTC + 255) / 256, blk256, 0, stream>>>(
          (const float2*)accF, b1, hbuf, N * COUTC);
    }
  }

  final_kernel<<<(numTiles + 3) / 4, blk128, 0, stream>>>(
      x2, resw2, (const float2*)accF, b2, (float2*)d_out, numTiles, N);
}